// PNet_21371757264949
// MI455X (gfx1250) — compile-verified
//
#include <hip/hip_runtime.h>

// ---------------------------------------------------------------------------
// PNet forward, fused single-pass kernel for gfx1250 (MI455X).
//  - 1 row per lane, 32 rows per wave, 256-thread blocks (8 waves).
//  - Tiny layers (<=8x6): scalar VALU, weights via uniform (s_load) accesses.
//  - m_down_l1/l2/l2_1/l2_2: V_WMMA_F32_16X16X4_F32, two 16-row tiles/wave.
//    B (weights) loaded once per layer, A (activations) once per tile,
//    bias folded into the C accumulator, all operand selects branchless
//    (WMMA requires EXEC==all-ones; no exec-diverging loads near WMMA).
//  - Coalesced global I/O staged through per-wave LDS slabs (stride 21 ->
//    conflict-free A reads and row reads).
// Input pointer order assumed: d_in[0]=x, then per LAYERS order: w, b.
// ---------------------------------------------------------------------------

typedef __attribute__((ext_vector_type(2))) float v2f;
typedef __attribute__((ext_vector_type(8))) float v8f;

#define NLAYER 36
#define SMSTRIDE 21

struct PNetArgs {
  const float* x;
  float*       out;
  const float* w[NLAYER];
  const float* b[NLAYER];
  int          nrows;
};

__device__ __forceinline__ float fast_tanh(float x) {
  // tanh(x) = 1 - 2/(exp(2x)+1); exact limits at +/-inf, ~1e-6 rel err.
  return 1.0f - 2.0f / (__expf(2.0f * x) + 1.0f);
}

// ---- scalar tiny dense layers (uniform weights -> scalar loads) -----------
template<int FI, int FO>
__device__ __forceinline__ void fc_relu(const float* __restrict__ w,
                                        const float* __restrict__ b,
                                        const float* in, float* out) {
#pragma unroll
  for (int o = 0; o < FO; ++o) {
    float a = b[o];
#pragma unroll
    for (int k = 0; k < FI; ++k) a = fmaf(w[o * FI + k], in[k], a);
    out[o] = fmaxf(a, 0.0f);
  }
}

template<int FI, int FO>
__device__ __forceinline__ void fc_tanh(const float* __restrict__ w,
                                        const float* __restrict__ b,
                                        const float* in, float* out) {
#pragma unroll
  for (int o = 0; o < FO; ++o) {
    float a = b[o];
#pragma unroll
    for (int k = 0; k < FI; ++k) a = fmaf(w[o * FI + k], in[k], a);
    out[o] = fast_tanh(a);
  }
}

template<int FI, int FO>
__device__ __forceinline__ void fc_lin(const float* __restrict__ w,
                                       const float* __restrict__ b,
                                       const float* in, float* out) {
#pragma unroll
  for (int o = 0; o < FO; ++o) {
    float a = b[o];
#pragma unroll
    for (int k = 0; k < FI; ++k) a = fmaf(w[o * FI + k], in[k], a);
    out[o] = a;
  }
}

// ---- WMMA operand containers ----------------------------------------------
// A layout (32-bit 16x4): VGPR0: lanes0-15 K=k0,   lanes16-31 K=k0+2
//                         VGPR1: lanes0-15 K=k0+1, lanes16-31 K=k0+3
// B layout (32-bit 4x16): VGPR0: lanes0-15 row K=k0 (N=lane), lanes16-31 K=k0+2
// C/D: VGPR v: lanes0-15 (M=v,N=lane), lanes16-31 (M=v+8,N=lane-16)
template<int K> struct AReg { v2f a[(K + 3) / 4]; };
template<int K> struct BReg { v2f b[(K + 3) / 4]; float bias; };

// Load B (= W^T chunks) + bias once per layer; fully branchless:
// clamp addresses to a legal index, mask invalid lanes with cndmask.
template<int K, int FO>
__device__ __forceinline__ BReg<K> load_B(int lane,
                                          const float* __restrict__ W,
                                          const float* __restrict__ Bb) {
  BReg<K> r;
  const int n  = lane & 15;
  const int hf = lane >> 4;
  const int wlim = K * FO - 1;
  const float bv = Bb[n < FO ? n : 0];
  r.bias = (n < FO) ? bv : 0.0f;
#pragma unroll
  for (int c = 0; c < (K + 3) / 4; ++c) {
    const int k0 = 4 * c + 2 * hf;
    const int i0 = n * K + k0;
    const int i1 = i0 + 1;
    const float w0 = W[i0 <= wlim ? i0 : 0];   // always-legal load
    const float w1 = W[i1 <= wlim ? i1 : 0];
    const bool ok0 = (n < FO) && (k0     < K);
    const bool ok1 = (n < FO) && (k0 + 1 < K);
    r.b[c].x = ok0 ? w0 : 0.0f;
    r.b[c].y = ok1 ? w1 : 0.0f;
  }
  return r;
}

// Load A chunks once per 16-row tile (LDS cols >= K hold zeros).
template<int K>
__device__ __forceinline__ AReg<K> load_A(const float* __restrict__ sA, int lane) {
  AReg<K> r;
  const int hf = lane >> 4;
  const float* ar = sA + (size_t)(lane & 15) * SMSTRIDE;
#pragma unroll
  for (int c = 0; c < (K + 3) / 4; ++c) {
    const int k0 = 4 * c + 2 * hf;
    r.a[c].x = ar[k0];
    r.a[c].y = ar[k0 + 1];
  }
  return r;
}

template<int K>
__device__ __forceinline__ v8f wmma_apply(const AReg<K>& A, const BReg<K>& B) {
  v8f acc;
#pragma unroll
  for (int v = 0; v < 8; ++v) acc[v] = B.bias;     // bias folded into C
#pragma unroll
  for (int c = 0; c < (K + 3) / 4; ++c)
    acc = __builtin_amdgcn_wmma_f32_16x16x4_f32(false, A.a[c], false, B.b[c],
                                                (short)0, acc, false, false);
  return acc;
}

__device__ __forceinline__ void store_relu_D(float* __restrict__ sD, int lane, v8f d) {
  const int n  = lane & 15;
  const int hf = lane >> 4;
#pragma unroll
  for (int v = 0; v < 8; ++v)
    sD[(size_t)(v + 8 * hf) * SMSTRIDE + n] = fmaxf(d[v], 0.0f);
}

__launch_bounds__(256, 2)
__global__ void pnet_fused(PNetArgs A) {
  __shared__ float sm[8][32 * SMSTRIDE];            // 672 f32 per wave, 21 KB
  const int tid  = threadIdx.x;
  const int wave = tid >> 5;
  const int lane = tid & 31;
  float* sw = sm[wave];

  const int waveRow0 = blockIdx.x * 256 + wave * 32;

  // ---- coalesced wave load: 32 rows x 17 f32 = 544 contiguous floats ----
  {
    const size_t lim = (size_t)A.nrows * 17 - 1;
    const size_t g0  = (size_t)waveRow0 * 17;
#pragma unroll
    for (int i = 0; i < 17; ++i) {
      size_t gi = g0 + lane + 32 * i;
      sw[lane + 32 * i] = A.x[gi <= lim ? gi : lim];
    }
  }
  __syncthreads();

  // ---- each lane pulls its own row ----
  float xi[17];
#pragma unroll
  for (int k = 0; k < 17; ++k) xi[k] = sw[lane * 17 + k];
  __syncthreads();

  const float h1 = xi[2],  k1v = xi[3],  f1 = xi[4];
  const float h2 = xi[5],  k2v = xi[6],  f2 = xi[7];
  const float h1d = xi[11], k1d = xi[12], f1d = xi[13];
  const float h2d = xi[14], k2d = xi[15], f2d = xi[16];

  // ---- scalar "up" chains -> h_up_1, h_up_2 ----
  float hu1[6], hu2[6];
  {
    float t0[2] = { f1, f1d }, t1[2], fu[2];
    fc_relu<2, 2>(A.w[0], A.b[0], t0, t1);
    fc_relu<2, 2>(A.w[1], A.b[1], t1, fu);
    float t2[4] = { k1v, k1d, fu[0], fu[1] }, t3[4], ku[4];
    fc_relu<4, 4>(A.w[2], A.b[2], t2, t3);
    fc_relu<4, 4>(A.w[3], A.b[3], t3, ku);
    float t4[6] = { h1, h1d, ku[0], ku[1], ku[2], ku[3] }, t5[6];
    fc_relu<6, 6>(A.w[4], A.b[4], t4, t5);
    fc_relu<6, 6>(A.w[5], A.b[5], t5, hu1);
  }
  {
    float t0[2] = { f2, f2d }, t1[2], fu[2];
    fc_relu<2, 2>(A.w[20], A.b[20], t0, t1);
    fc_relu<2, 2>(A.w[21], A.b[21], t1, fu);
    float t2[4] = { k2v, k2d, fu[0], fu[1] }, t3[4], ku[4];
    fc_relu<4, 4>(A.w[22], A.b[22], t2, t3);
    fc_relu<4, 4>(A.w[23], A.b[23], t3, ku);
    float t4[6] = { h2, h2d, ku[0], ku[1], ku[2], ku[3] }, t5[6];
    fc_relu<6, 6>(A.w[24], A.b[24], t4, t5);
    fc_relu<6, 6>(A.w[25], A.b[25], t5, hu2);
  }

  // ---- write 17-feature m_down input (+ zero pad to K=20) ----
  {
    float* r = sw + (size_t)lane * SMSTRIDE;
    r[0] = xi[0]; r[1] = xi[1]; r[2] = xi[8]; r[3] = xi[9]; r[4] = xi[10];
#pragma unroll
    for (int j = 0; j < 6; ++j) { r[5 + j] = hu1[j]; r[11 + j] = hu2[j]; }
    r[17] = 0.0f; r[18] = 0.0f; r[19] = 0.0f;
  }
  __syncthreads();

  // ---- WMMA m_down stack: B once per layer, two 16-row tiles per wave ----
  {                                                 // m_down_l1: 17 -> 16
    const BReg<17> B = load_B<17, 16>(lane, A.w[6], A.b[6]);
#pragma unroll
    for (int t = 0; t < 2; ++t) {
      float* sT = sw + (size_t)(t * 16) * SMSTRIDE;
      const AReg<17> Ar = load_A<17>(sT, lane);
      store_relu_D(sT, lane, wmma_apply<17>(Ar, B));
    }
  }
  {                                                 // m_down_l2: 16 -> 16
    const BReg<16> B = load_B<16, 16>(lane, A.w[7], A.b[7]);
#pragma unroll
    for (int t = 0; t < 2; ++t) {
      float* sT = sw + (size_t)(t * 16) * SMSTRIDE;
      const AReg<16> Ar = load_A<16>(sT, lane);
      store_relu_D(sT, lane, wmma_apply<16>(Ar, B));
    }
  }
  {                                                 // m_down_l2_1 / l2_2: 16 -> 6
    const BReg<16> B1 = load_B<16, 6>(lane, A.w[8], A.b[8]);
    const BReg<16> B2 = load_B<16, 6>(lane, A.w[9], A.b[9]);
    const int n = lane & 15, hf = lane >> 4;
#pragma unroll
    for (int t = 0; t < 2; ++t) {
      float* sT = sw + (size_t)(t * 16) * SMSTRIDE;
      const AReg<16> Ar = load_A<16>(sT, lane);     // shared by both layers
      const v8f d1 = wmma_apply<16>(Ar, B1);
      const v8f d2 = wmma_apply<16>(Ar, B2);
      if (n < 6) {                                  // divergence OK: no WMMA here
#pragma unroll
        for (int v = 0; v < 8; ++v) {
          float* rp = sT + (size_t)(v + 8 * hf) * SMSTRIDE;
          rp[n]     = fmaxf(d1[v], 0.0f);           // m_down_1 -> cols 0..5
          rp[8 + n] = fmaxf(d2[v], 0.0f);           // m_down_2 -> cols 8..13
        }
      }
    }
  }
  __syncthreads();

  // ---- read back m_down_1 / m_down_2 for this lane's row ----
  float md1[6], md2[6];
  {
    const float* r = sw + (size_t)lane * SMSTRIDE;
#pragma unroll
    for (int c = 0; c < 6; ++c) { md1[c] = r[c]; md2[c] = r[8 + c]; }
  }

  // ---- scalar "down"/"act" chains ----
  float h_act_1, k_act_1, f_act_1, h_act_2, k_act_2, f_act_2;
  {
    float hdt[6], hd[6], kdt[4], kd[4];
    fc_tanh<6, 6>(A.w[10], A.b[10], md1, hdt);
    fc_relu<6, 6>(A.w[11], A.b[11], hdt, hd);
    fc_tanh<6, 4>(A.w[12], A.b[12], hd, kdt);
    fc_relu<4, 4>(A.w[13], A.b[13], kdt, kd);
    float hin[8] = { h1, h1d, md1[0], md1[1], md1[2], md1[3], md1[4], md1[5] }, ht[4];
    fc_relu<8, 4>(A.w[14], A.b[14], hin, ht);
    fc_lin<4, 1>(A.w[15], A.b[15], ht, &h_act_1);
    float kin[8] = { k1v, k1d, hd[0], hd[1], hd[2], hd[3], hd[4], hd[5] }, kt[4];
    fc_relu<8, 4>(A.w[16], A.b[16], kin, kt);
    fc_lin<4, 1>(A.w[17], A.b[17], kt, &k_act_1);
    float fin[6] = { f1, f1d, kd[0], kd[1], kd[2], kd[3] }, ft[4];
    fc_relu<6, 4>(A.w[18], A.b[18], fin, ft);
    fc_lin<4, 1>(A.w[19], A.b[19], ft, &f_act_1);
  }
  {
    float hdt[6], hd[6], kdt[4], kd[4];
    fc_tanh<6, 6>(A.w[26], A.b[26], md2, hdt);
    fc_relu<6, 6>(A.w[27], A.b[27], hdt, hd);
    fc_tanh<6, 4>(A.w[28], A.b[28], hd, kdt);
    fc_relu<4, 4>(A.w[29], A.b[29], kdt, kd);
    float hin[8] = { h2, h2d, md2[0], md2[1], md2[2], md2[3], md2[4], md2[5] }, ht[4];
    fc_relu<8, 4>(A.w[30], A.b[30], hin, ht);
    fc_lin<4, 1>(A.w[31], A.b[31], ht, &h_act_2);
    float kin[8] = { k2v, k2d, hd[0], hd[1], hd[2], hd[3], hd[4], hd[5] }, kt[4];
    fc_relu<8, 4>(A.w[32], A.b[32], kin, kt);
    fc_lin<4, 1>(A.w[33], A.b[33], kt, &k_act_2);
    float fin[6] = { f2, f2d, kd[0], kd[1], kd[2], kd[3] }, ft[4];
    fc_relu<6, 4>(A.w[34], A.b[34], fin, ft);
    fc_lin<4, 1>(A.w[35], A.b[35], ft, &f_act_2);
  }

  __syncthreads();
  // ---- stage outputs, then coalesced wave store (192 contiguous f32) ----
  {
    float* r = sw + (size_t)lane * 6;
    r[0] = h_act_1; r[1] = f_act_1; r[2] = k_act_1;   // leg1: h, f, k
    r[3] = h_act_2; r[4] = k_act_2; r[5] = f_act_2;   // leg2: h, k, f
  }
  __syncthreads();
  {
    const size_t o0   = (size_t)waveRow0 * 6;
    const size_t olim = (size_t)A.nrows * 6;
#pragma unroll
    for (int i = 0; i < 6; ++i) {
      size_t oi = o0 + lane + 32 * i;
      if (oi < olim) A.out[oi] = sw[lane + 32 * i];
    }
  }
}

extern "C" void kernel_launch(void* const* d_in, const int* in_sizes, int n_in,
                              void* d_out, int out_size, void* d_ws, size_t ws_size,
                              hipStream_t stream) {
  (void)d_ws; (void)ws_size; (void)out_size; (void)n_in;
  PNetArgs a;
  a.x     = (const float*)d_in[0];
  a.out   = (float*)d_out;
  a.nrows = in_sizes[0] / 17;
  for (int i = 0; i < NLAYER; ++i) {
    a.w[i] = (const float*)d_in[1 + 2 * i];
    a.b[i] = (const float*)d_in[2 + 2 * i];
  }
  const int nblocks = (a.nrows + 255) / 256;        // B=2^21 -> 8192 blocks
  pnet_fused<<<nblocks, 256, 0, stream>>>(a);
}